// MotorSchemaGrounder_87351044866601
// MI455X (gfx1250) — compile-verified
//
#include <hip/hip_runtime.h>
#include <cmath>

typedef float v2f __attribute__((ext_vector_type(2)));
typedef float v8f __attribute__((ext_vector_type(8)));

#define KC 64

__device__ __forceinline__ float sigm(float x) { return 1.0f / (1.0f + __expf(-x)); }
__device__ __forceinline__ float gelu_exact(float x) { return 0.5f * x * (1.0f + erff(x * 0.70710678118654752f)); }

// async global->LDS 16-byte copy (CDNA5): vdst = LDS byte offset, vaddr = 64-bit global.
// LDS generic pointers keep the workgroup-LDS byte offset in the low 32 bits.
__device__ __forceinline__ void async_ld_b128(const float* g, const float* l)
{
    asm volatile("global_load_async_to_lds_b128 %0, %1, off"
                 :: "v"((unsigned)(size_t)l), "v"((unsigned long long)(size_t)g)
                 : "memory");
}

// ---------------------------------------------------------------------------
// Generic 16xN = (16xK) @ (KxN) wave-level WMMA GEMM (f32, 16x16x4).
// TB=false: B is KxN row-major.  TB=true: B is NxK row-major (computes A@B^T).
// Block = 4 waves = 64 output columns; K consumed in 64-wide chunks that are
// double-buffered through LDS with global_load_async_to_lds_b128 (ASYNCcnt),
// so chunk i+1 streams in while chunk i feeds 16 V_WMMA_F32_16X16X4_F32 ops.
// Only rows 0..7 of A/C are meaningful (M=8 padded to 16).
// ---------------------------------------------------------------------------
template <bool TB>
__global__ __launch_bounds__(128) void k_gemm16(
    const float* __restrict__ A, long sA,
    const float* __restrict__ B, long sB,
    float* __restrict__ C, long sC,
    int K, int N)
{
    A += sA * blockIdx.y;
    B += sB * blockIdx.y;
    C += sC * blockIdx.y;

    constexpr int ASTR = 68;              // 16B-aligned rows, conflict-free frags
    constexpr int BSTR = TB ? 68 : 72;
    __shared__ __align__(16) float AS[2][16 * ASTR];
    __shared__ __align__(16) float BS[2][64 * BSTR];

    const int tid  = threadIdx.x;
    const int lane = tid & 31;
    const int wave = tid >> 5;
    const int nblk = blockIdx.x * 64;
    const int row  = lane & 15;           // M for A-frag / N for B-frag
    const int kh   = (lane >> 4) << 1;    // 0 or 2 (K half select)
    const int nn   = nblk + wave * 16 + row;

    auto prefetch = [&](int kc, int bufi) {
        // B chunk: 64 rows x 64 floats (rows = k for KxN, rows = n for NxK)
        #pragma unroll
        for (int j = 0; j < 8; ++j) {
            int t  = tid + j * 128;
            int rr = t >> 4;
            int c4 = (t & 15) << 2;
            const float* g = TB ? (B + (long)(nblk + rr) * K + kc + c4)
                                : (B + (long)(kc + rr) * N + nblk + c4);
            async_ld_b128(g, &BS[bufi][rr * BSTR + c4]);
        }
        // A chunk: 16 rows x 64 floats
        #pragma unroll
        for (int j = 0; j < 2; ++j) {
            int t  = tid + j * 128;
            int m  = t >> 4;
            int c4 = (t & 15) << 2;
            async_ld_b128(A + (long)m * K + kc + c4, &AS[bufi][m * ASTR + c4]);
        }
    };

    v8f acc = {0.f, 0.f, 0.f, 0.f, 0.f, 0.f, 0.f, 0.f};

    prefetch(0, 0);
    const int nchunks = K / KC;
    for (int it = 0; it < nchunks; ++it) {
        const int cur = it & 1;
        if (it + 1 < nchunks) {
            prefetch((it + 1) * KC, cur ^ 1);
            asm volatile("s_wait_asynccnt 0xa" ::: "memory");  // current chunk landed
        } else {
            asm volatile("s_wait_asynccnt 0x0" ::: "memory");
        }
        __syncthreads();

        const float* as = AS[cur];
        const float* bs = BS[cur];
        #pragma unroll
        for (int kk = 0; kk < KC; kk += 4) {
            v2f a, b;
            a.x = as[row * ASTR + kk + kh];
            a.y = as[row * ASTR + kk + kh + 1];
            if (TB) {
                int bi = (wave * 16 + row) * BSTR + kk + kh;
                b.x = bs[bi];
                b.y = bs[bi + 1];
            } else {
                int bi = (kk + kh) * BSTR + wave * 16 + row;
                b.x = bs[bi];
                b.y = bs[bi + BSTR];
            }
            acc = __builtin_amdgcn_wmma_f32_16x16x4_f32(
                false, a, false, b, (short)0, acc, false, false);
        }
        __syncthreads();
    }

    // C/D layout: VGPR r -> M = r + (lane/16)*8, N = lane%16
    const int mb = (lane >> 4) << 3;
    #pragma unroll
    for (int r = 0; r < 8; ++r)
        C[(long)(mb + r) * N + nn] = acc[r];
}

// ---------------------------------------------------------------------------
// pooled = mean over L of x (split L into 8 slices; float4-wide streaming)
// ---------------------------------------------------------------------------
__global__ __launch_bounds__(256) void k_pool_part(const float* __restrict__ x,
                                                   float* __restrict__ part)
{
    int dq = blockIdx.x * 256 + threadIdx.x;   // float4 column, grid.x = 2
    int b = blockIdx.y, ls = blockIdx.z;
    const float4* xb = (const float4*)(x + ((size_t)b * 2048 + (size_t)ls * 256) * 2048);
    float4 acc = {0.f, 0.f, 0.f, 0.f};
    for (int l = 0; l < 256; ++l) {
        float4 v = xb[(size_t)l * 512 + dq];
        acc.x += v.x; acc.y += v.y; acc.z += v.z; acc.w += v.w;
    }
    ((float4*)part)[((size_t)ls * 8 + b) * 512 + dq] = acc;
}

__global__ __launch_bounds__(256) void k_pool_reduce(const float* __restrict__ part,
                                                     float* __restrict__ pooled)
{
    int i = blockIdx.x * 256 + threadIdx.x;    // 16*2048, grid 128
    int b = i >> 11, dd = i & 2047;
    float v = 0.f;
    if (b < 8) {
        float acc = 0.f;
        for (int z = 0; z < 8; ++z) acc += part[((size_t)z * 8 + b) * 2048 + dd];
        v = acc * (1.f / 2048.f);
    }
    pooled[i] = v;   // rows 8..15 zero-padded
}

// ---------------------------------------------------------------------------
// embh[p][h] = prim_emb[p] @ pre_w1[p, 2048:, :]   (split K for parallelism)
// ---------------------------------------------------------------------------
__global__ __launch_bounds__(128) void k_emb_part(const float* __restrict__ prim_emb,
                                                  const float* __restrict__ pre_w1,
                                                  float* __restrict__ part)
{
    int p = blockIdx.x, ks = blockIdx.y;       // grid(32,16), block 128
    int h = threadIdx.x;
    const float* e  = prim_emb + (size_t)p * 2048 + (size_t)ks * 128;
    const float* wv = pre_w1 + (size_t)p * 4096 * 128 + (size_t)(2048 + ks * 128) * 128;
    float acc = 0.f;
    for (int k = 0; k < 128; ++k) acc += e[k] * wv[(size_t)k * 128 + h];
    part[((size_t)ks * 32 + p) * 128 + h] = acc;
}

__global__ __launch_bounds__(256) void k_emb_reduce(const float* __restrict__ part,
                                                    float* __restrict__ embh)
{
    int i = blockIdx.x * 256 + threadIdx.x;    // 32*128 = 4096, grid 16
    int p = i >> 7, h = i & 127;
    float acc = 0.f;
    for (int z = 0; z < 16; ++z) acc += part[((size_t)z * 32 + p) * 128 + h];
    embh[i] = acc;
}

// ---------------------------------------------------------------------------
// elementwise epilogues on padded 16xN buffers (rows >= 8 forced to zero)
// ---------------------------------------------------------------------------
__global__ __launch_bounds__(256) void k_bias_gelu(
    const float* __restrict__ in, long sIn, float* __restrict__ out, long sOut,
    const float* __restrict__ b0, long sB0, const float* __restrict__ b1, long sB1, int N)
{
    in  += sIn  * blockIdx.y;
    out += sOut * blockIdx.y;
    int i = blockIdx.x * 256 + threadIdx.x;
    if (i >= 16 * N) return;
    int r = i / N, c = i % N;
    float v = 0.f;
    if (r < 8) {
        v = in[i];
        if (b0) v += b0[sB0 * blockIdx.y + c];
        if (b1) v += b1[sB1 * blockIdx.y + c];
        v = gelu_exact(v);
    }
    out[i] = v;
}

__global__ __launch_bounds__(256) void k_bias_add(
    float* __restrict__ data, long sD, const float* __restrict__ b0, long sB, int N)
{
    data += sD * blockIdx.y;
    int i = blockIdx.x * 256 + threadIdx.x;
    if (i >= 16 * N) return;
    int r = i / N, c = i % N;
    data[i] = (r < 8) ? data[i] + b0[sB * blockIdx.y + c] : 0.f;
}

// ---------------------------------------------------------------------------
// selector: logits = s @ sel_w2 + b2, softmax over P=32
// ---------------------------------------------------------------------------
__global__ __launch_bounds__(256) void k_sel_softmax(
    const float* __restrict__ s, const float* __restrict__ w2,
    const float* __restrict__ b2, float* __restrict__ prim)
{
    __shared__ float lg[8][32];
    int t = threadIdx.x;               // 256 threads -> (b,p)
    int b = t >> 5, p = t & 31;
    float acc = b2[p];
    for (int h = 0; h < 128; ++h) acc += s[b * 128 + h] * w2[h * 32 + p];
    lg[b][p] = acc;
    __syncthreads();
    if (t < 8) {
        float mx = lg[t][0];
        for (int q = 1; q < 32; ++q) mx = fmaxf(mx, lg[t][q]);
        float e[32], sum = 0.f;
        for (int q = 0; q < 32; ++q) { e[q] = __expf(lg[t][q] - mx); sum += e[q]; }
        for (int q = 0; q < 32; ++q) prim[t * 32 + q] = e[q] / sum;
    }
}

// exec_tensor[b][p] = sigmoid(<h_pre[b,p,:], pre_w2[p,:]> + pre_b2[p])
__global__ __launch_bounds__(256) void k_exec(
    const float* __restrict__ hpre, const float* __restrict__ pre_w2,
    const float* __restrict__ pre_b2, float* __restrict__ ex)
{
    int t = threadIdx.x;
    int b = t >> 5, p = t & 31;
    float acc = pre_b2[p];
    for (int h = 0; h < 128; ++h) acc += hpre[((size_t)p * 16 + b) * 128 + h] * pre_w2[p * 128 + h];
    ex[b * 32 + p] = sigm(acc);
}

// top-3 of prim*exec per batch row (ties -> lowest index, matching lax.top_k)
__global__ void k_topk(const float* __restrict__ prim, const float* __restrict__ ex,
                       int* __restrict__ tidx)
{
    int b = threadIdx.x;
    if (b < 8) {
        float f[32];
        for (int p = 0; p < 32; ++p) f[p] = prim[b * 32 + p] * ex[b * 32 + p];
        for (int s = 0; s < 3; ++s) {
            int bi = 0; float bv = f[0];
            for (int p = 1; p < 32; ++p) if (f[p] > bv) { bv = f[p]; bi = p; }
            tidx[b * 3 + s] = bi;
            f[bi] = -1e30f;
        }
    }
}

__global__ __launch_bounds__(256) void k_zero(float* __restrict__ p, int n)
{
    int i = blockIdx.x * 256 + threadIdx.x;
    if (i < n) p[i] = 0.f;
}

// gather selected effect vector for this step into padded 16x256 A operand
__global__ __launch_bounds__(256) void k_gather(
    const float* __restrict__ effect, const int* __restrict__ tidx, int step,
    float* __restrict__ effsel)
{
    int i = blockIdx.x * 256 + threadIdx.x;   // 16*256, grid 16
    int b = i >> 8, a = i & 255;
    float v = 0.f;
    if (b < 8) {
        int p = tidx[b * 3 + step];
        v = effect[((size_t)p * 16 + b) * 256 + a];
    }
    effsel[i] = v;
}

// GRU gate math + state update (biases folded here)
__global__ __launch_bounds__(256) void k_gru(
    const float* __restrict__ gi, const float* __restrict__ gh,
    const float* __restrict__ bi, const float* __restrict__ bh,
    float* __restrict__ h)
{
    int i = blockIdx.x * 256 + threadIdx.x;   // 8*2048, grid 64
    int b = i >> 11, d = i & 2047;
    const float* gib = gi + (size_t)b * 6144;
    const float* ghb = gh + (size_t)b * 6144;
    float gir = gib[d]        + bi[d];
    float giz = gib[2048 + d] + bi[2048 + d];
    float gin = gib[4096 + d] + bi[4096 + d];
    float ghr = ghb[d]        + bh[d];
    float ghz = ghb[2048 + d] + bh[2048 + d];
    float ghn = ghb[4096 + d] + bh[4096 + d];
    float r = sigm(gir + ghr);
    float z = sigm(giz + ghz);
    float n = tanhf(gin + r * ghn);
    float ho = h[(size_t)b * 2048 + d];
    h[(size_t)b * 2048 + d] = (1.f - z) * n + z * ho;
}

// per-row rmsnorm of (in + bias) with weight w, over padded 16xD buffer
__global__ __launch_bounds__(256) void k_rms8(
    const float* __restrict__ in, const float* __restrict__ bias,
    const float* __restrict__ w, float* __restrict__ out, int D)
{
    int b = blockIdx.x;   // 0..15
    __shared__ float red[256];
    float acc = 0.f;
    if (b < 8)
        for (int d = threadIdx.x; d < D; d += 256) {
            float v = in[(size_t)b * D + d] + bias[d];
            acc += v * v;
        }
    red[threadIdx.x] = acc;
    __syncthreads();
    for (int s = 128; s > 0; s >>= 1) {
        if (threadIdx.x < s) red[threadIdx.x] += red[threadIdx.x + s];
        __syncthreads();
    }
    float rs = rsqrtf(red[0] / (float)D + 1e-6f);
    for (int d = threadIdx.x; d < D; d += 256) {
        float v = 0.f;
        if (b < 8) v = (in[(size_t)b * D + d] + bias[d]) * rs * w[d];
        out[(size_t)b * D + d] = v;
    }
}

// g = silu(u1) * u2 on padded 16x4096
__global__ __launch_bounds__(256) void k_swiglu(
    const float* __restrict__ u1, const float* __restrict__ u2, float* __restrict__ g)
{
    int i = blockIdx.x * 256 + threadIdx.x;   // 16*4096, grid 256
    int r = i >> 12;
    float v = 0.f;
    if (r < 8) {
        float a = u1[i];
        v = a * sigm(a) * u2[i];
    }
    g[i] = v;
}

// out = rmsnorm(x + 0.1*motor_lang, norm_w) per (b,l) row, float4-wide
__global__ __launch_bounds__(256) void k_final(
    const float* __restrict__ x, const float* __restrict__ ml,
    const float* __restrict__ nw, float* __restrict__ out)
{
    int rowi = blockIdx.x;             // 8*2048 rows
    int b = rowi >> 11;
    const float4* xr  = (const float4*)(x  + (size_t)rowi * 2048);
    const float4* mb  = (const float4*)(ml + (size_t)b * 2048);
    const float4* nw4 = (const float4*)nw;
    float4* o4 = (float4*)(out + (size_t)rowi * 2048);
    __shared__ float red[256];

    const int i0 = threadIdx.x, i1 = threadIdx.x + 256;   // 512 float4 per row
    float4 a0 = xr[i0], m0 = mb[i0];
    float4 a1 = xr[i1], m1 = mb[i1];
    float4 v0 = {a0.x + 0.1f * m0.x, a0.y + 0.1f * m0.y, a0.z + 0.1f * m0.z, a0.w + 0.1f * m0.w};
    float4 v1 = {a1.x + 0.1f * m1.x, a1.y + 0.1f * m1.y, a1.z + 0.1f * m1.z, a1.w + 0.1f * m1.w};
    float acc = v0.x*v0.x + v0.y*v0.y + v0.z*v0.z + v0.w*v0.w
              + v1.x*v1.x + v1.y*v1.y + v1.z*v1.z + v1.w*v1.w;

    red[threadIdx.x] = acc;
    __syncthreads();
    for (int s = 128; s > 0; s >>= 1) {
        if (threadIdx.x < s) red[threadIdx.x] += red[threadIdx.x + s];
        __syncthreads();
    }
    float rs = rsqrtf(red[0] * (1.f / 2048.f) + 1e-6f);

    float4 w0 = nw4[i0], w1 = nw4[i1];
    float4 r0 = {v0.x*rs*w0.x, v0.y*rs*w0.y, v0.z*rs*w0.z, v0.w*rs*w0.w};
    float4 r1 = {v1.x*rs*w1.x, v1.y*rs*w1.y, v1.z*rs*w1.z, v1.w*rs*w1.w};
    o4[i0] = r0;
    o4[i1] = r1;
}

// ---------------------------------------------------------------------------
extern "C" void kernel_launch(void* const* d_in, const int* in_sizes, int n_in,
                              void* d_out, int out_size, void* d_ws, size_t ws_size,
                              hipStream_t stream)
{
    (void)in_sizes; (void)n_in; (void)out_size; (void)ws_size;
    const float* x        = (const float*)d_in[0];
    const float* prim_emb = (const float*)d_in[1];
    const float* pre_w1   = (const float*)d_in[2];
    const float* pre_b1   = (const float*)d_in[3];
    const float* pre_w2   = (const float*)d_in[4];
    const float* pre_b2   = (const float*)d_in[5];
    const float* eff_w1   = (const float*)d_in[6];
    const float* eff_b1   = (const float*)d_in[7];
    const float* eff_w2   = (const float*)d_in[8];
    const float* eff_b2   = (const float*)d_in[9];
    const float* sel_w1   = (const float*)d_in[10];
    const float* sel_b1   = (const float*)d_in[11];
    const float* sel_w2   = (const float*)d_in[12];
    const float* sel_b2   = (const float*)d_in[13];
    const float* gru_wi   = (const float*)d_in[14];
    const float* gru_bi   = (const float*)d_in[15];
    const float* gru_wh   = (const float*)d_in[16];
    const float* gru_bh   = (const float*)d_in[17];
    const float* m2l_w1   = (const float*)d_in[18];
    const float* m2l_b1   = (const float*)d_in[19];
    const float* m2l_rms  = (const float*)d_in[20];
    const float* sw_w1    = (const float*)d_in[21];
    const float* sw_w2    = (const float*)d_in[22];
    const float* sw_w3    = (const float*)d_in[23];
    const float* m2l_w2   = (const float*)d_in[24];
    const float* m2l_b2   = (const float*)d_in[25];
    const float* norm_w   = (const float*)d_in[26];
    float* out = (float*)d_out;

    float* w = (float*)d_ws;
    size_t o = 0;
    auto alloc = [&](size_t n) { float* p = w + o; o += n; return p; };

    float* poolpart = alloc(8 * 8 * 2048);
    float* pooled   = alloc(16 * 2048);
    float* embpart  = alloc(16 * 32 * 128);
    float* embh     = alloc(32 * 128);
    float* sbuf     = alloc(16 * 128);
    float* prim     = alloc(8 * 32);
    float* hpre     = alloc(32 * 16 * 128);
    float* exbuf    = alloc(8 * 32);
    float* heff     = alloc(32 * 16 * 128);
    float* effect   = alloc(32 * 16 * 256);
    int*   tidx     = (int*)alloc(32);
    float* effsel   = alloc(16 * 256);
    float* hpad     = alloc(16 * 2048);
    float* gi       = alloc(16 * 6144);
    float* gh       = alloc(16 * 6144);
    float* mraw     = alloc(16 * 2048);
    float* mn       = alloc(16 * 2048);
    float* u1       = alloc(16 * 4096);
    float* u2       = alloc(16 * 4096);
    float* gpad     = alloc(16 * 4096);
    float* swr      = alloc(16 * 2048);
    float* ml       = alloc(16 * 2048);

    const float* nullf = nullptr;

    // pooled = mean_L(x), zero-padded to 16 rows
    k_pool_part<<<dim3(2, 8, 8), 256, 0, stream>>>(x, poolpart);
    k_pool_reduce<<<128, 256, 0, stream>>>(poolpart, pooled);

    // embh[p] = prim_emb[p] @ w_emb[p]
    k_emb_part<<<dim3(32, 16), 128, 0, stream>>>(prim_emb, pre_w1, embpart);
    k_emb_reduce<<<16, 256, 0, stream>>>(embpart, embh);

    // selector: s = gelu(pooled @ sel_w1 + b1); prim = softmax(s @ sel_w2 + b2)
    k_gemm16<false><<<dim3(2, 1), 128, 0, stream>>>(pooled, 0, sel_w1, 0, sbuf, 0, 2048, 128);
    k_bias_gelu<<<dim3(8, 1), 256, 0, stream>>>(sbuf, 0, sbuf, 0, sel_b1, 0, nullf, 0, 128);
    k_sel_softmax<<<1, 256, 0, stream>>>(sbuf, sel_w2, sel_b2, prim);

    // h_pre = gelu(pooled @ w_state[p] + embh[p] + pre_b1[p]); exec = sigmoid(<h_pre, pre_w2>)
    k_gemm16<false><<<dim3(2, 32), 128, 0, stream>>>(pooled, 0, pre_w1, (long)4096 * 128, hpre, 16 * 128, 2048, 128);
    k_bias_gelu<<<dim3(8, 32), 256, 0, stream>>>(hpre, 16 * 128, hpre, 16 * 128, pre_b1, 128, embh, 128, 128);
    k_exec<<<1, 256, 0, stream>>>(hpre, pre_w2, pre_b2, exbuf);

    // h_eff = gelu(pooled @ eff_w1[p] + eff_b1[p]); effect = h_eff @ eff_w2[p] + eff_b2[p]
    k_gemm16<false><<<dim3(2, 32), 128, 0, stream>>>(pooled, 0, eff_w1, (long)2048 * 128, heff, 16 * 128, 2048, 128);
    k_bias_gelu<<<dim3(8, 32), 256, 0, stream>>>(heff, 16 * 128, heff, 16 * 128, eff_b1, 128, nullf, 0, 128);
    k_gemm16<false><<<dim3(4, 32), 128, 0, stream>>>(heff, 16 * 128, eff_w2, 128 * 256, effect, 16 * 256, 128, 256);
    k_bias_add<<<dim3(16, 32), 256, 0, stream>>>(effect, 16 * 256, eff_b2, 256, 256);

    // top-3 primitives per batch row
    k_topk<<<1, 32, 0, stream>>>(prim, exbuf, tidx);

    // GRU over K=3 selected primitives
    k_zero<<<128, 256, 0, stream>>>(hpad, 16 * 2048);
    for (int step = 0; step < 3; ++step) {
        k_gather<<<16, 256, 0, stream>>>(effect, tidx, step, effsel);
        // gi = eff @ gru_wi^T   (B given as N x K = 6144 x 256)
        k_gemm16<true><<<dim3(96, 1), 128, 0, stream>>>(effsel, 0, gru_wi, 0, gi, 0, 256, 6144);
        // gh = h @ gru_wh^T     (B given as N x K = 6144 x 2048)
        k_gemm16<true><<<dim3(96, 1), 128, 0, stream>>>(hpad, 0, gru_wh, 0, gh, 0, 2048, 6144);
        k_gru<<<64, 256, 0, stream>>>(gi, gh, gru_bi, gru_bh, hpad);
    }

    // motor-to-language head
    k_gemm16<false><<<dim3(32, 1), 128, 0, stream>>>(hpad, 0, m2l_w1, 0, mraw, 0, 2048, 2048);
    k_rms8<<<16, 256, 0, stream>>>(mraw, m2l_b1, m2l_rms, mn, 2048);
    k_gemm16<false><<<dim3(64, 1), 128, 0, stream>>>(mn, 0, sw_w1, 0, u1, 0, 2048, 4096);
    k_gemm16<false><<<dim3(64, 1), 128, 0, stream>>>(mn, 0, sw_w2, 0, u2, 0, 2048, 4096);
    k_swiglu<<<256, 256, 0, stream>>>(u1, u2, gpad);
    k_gemm16<false><<<dim3(32, 1), 128, 0, stream>>>(gpad, 0, sw_w3, 0, swr, 0, 4096, 2048);
    k_gemm16<false><<<dim3(32, 1), 128, 0, stream>>>(swr, 0, m2l_w2, 0, ml, 0, 2048, 2048);
    k_bias_add<<<dim3(128, 1), 256, 0, stream>>>(ml, 0, m2l_b2, 0, 2048);

    // x_motor = rmsnorm(x + 0.1*motor_lang, norm_w)
    k_final<<<16384, 256, 0, stream>>>(x, ml, norm_w, out);
}